// ModelIntegrator_6038724018483
// MI455X (gfx1250) — compile-verified
//
#include <hip/hip_runtime.h>
#include <stdint.h>

#define NHIST 6
#define NSIZE 8
#define NTIME 512
#define BLOCK 256
#define NEWTON_ITERS 3
#define NPHASE 4            // 4-deep async pipeline, prefetch distance 3

typedef float v4f __attribute__((ext_vector_type(4)));

__device__ __forceinline__ float frcp(float x) { return __builtin_amdgcn_rcpf(x); }

#define ASYNC_WAIT(N) asm volatile("s_wait_asynccnt " #N ::: "memory")

__global__ __launch_bounds__(BLOCK)
void chaboche_wb_kernel(const float* __restrict__ times,
                        const float* __restrict__ strains,
                        const float* __restrict__ temps,
                        const float* __restrict__ params,
                        float* __restrict__ out,
                        int nbatch)
{
    // 4-phase staging buffers for (time, strain, temp) of upcoming steps.
    // Wave-private slots -> no barriers, only ASYNCcnt ordering.
    __shared__ float sh_t[NPHASE][BLOCK];
    __shared__ float sh_e[NPHASE][BLOCK];
    __shared__ float sh_T[NPHASE][BLOCK];

    const int tid = threadIdx.x;
    int b = blockIdx.x * BLOCK + tid;
    if (b >= nbatch) b = nbatch - 1;   // clamped lanes duplicate the last sample (same data)

    // ---- material parameters (uniform) ----
    const float E0      = params[0];
    const float n       = params[1];
    const float inv_eta = frcp(params[2]);
    const float s0      = params[3];
    float C[NHIST], g[NHIST];
#pragma unroll
    for (int i = 0; i < NHIST; ++i) C[i] = params[4 + i];
#pragma unroll
    for (int i = 0; i < NHIST; ++i) g[i] = params[4 + NHIST + i];
    const float A   = params[4 + 2 * NHIST];
    const float nm1 = n - 1.0f;

    float y[NSIZE];
#pragma unroll
    for (int k = 0; k < NSIZE; ++k) y[k] = 0.0f;

    // t = 0 trajectory row: zero initial state (streaming NT store)
    {
        v4f z = (v4f)0.0f;
        v4f* op = (v4f*)(out + (size_t)b * NSIZE);
        __builtin_nontemporal_store(z, op);
        __builtin_nontemporal_store(z, op + 1);
    }

    float pt = times[b];    // previous time
    float pe = strains[b];  // previous strain

    // CDNA5 async copy global -> LDS (ASYNCcnt, no VGPR dest, no LOADcnt)
    auto issue_async = [&](int t, int ph) {
        const size_t off = (size_t)t * (size_t)nbatch + (size_t)b;
        uint32_t lt = (uint32_t)(uintptr_t)&sh_t[ph][tid];
        uint32_t le = (uint32_t)(uintptr_t)&sh_e[ph][tid];
        uint32_t lT = (uint32_t)(uintptr_t)&sh_T[ph][tid];
        asm volatile("global_load_async_to_lds_b32 %0, %1, off"
                     :: "v"(lt), "v"((uint64_t)(uintptr_t)(times + off)) : "memory");
        asm volatile("global_load_async_to_lds_b32 %0, %1, off"
                     :: "v"(le), "v"((uint64_t)(uintptr_t)(strains + off)) : "memory");
        asm volatile("global_load_async_to_lds_b32 %0, %1, off"
                     :: "v"(lT), "v"((uint64_t)(uintptr_t)(temps + off)) : "memory");
    };

    // Prime the pipeline: steps 1..3 in flight (phases 1,2,3)
    issue_async(1, 1);
    issue_async(2, 2);
    issue_async(3, 3);

#pragma unroll 1
    for (int t = 1; t < NTIME; ++t) {
        // keep prefetch distance 3: issue step t+3 before consuming step t
        if (t + 3 < NTIME) issue_async(t + 3, (t + 3) & (NPHASE - 1));

        // Async loads complete in order; wait until only the trios for
        // steps > t remain outstanding.  (scalar, uniform-t tail ladder)
        const int ahead = (NTIME - 1) - t;
        if      (ahead >= 3) ASYNC_WAIT(9);
        else if (ahead == 2) ASYNC_WAIT(6);
        else if (ahead == 1) ASYNC_WAIT(3);
        else                 ASYNC_WAIT(0);

        const int ph = t & (NPHASE - 1);
        const float ct = sh_t[ph][tid];
        const float ce = sh_e[ph][tid];
        const float T  = sh_T[ph][tid];

        const float dt = ct - pt;
        float er = (ce - pe) * frcp(dt);
        if (er != er) er = 0.0f;          // jnp.where(isnan(sr), 0, sr)
        pt = ct; pe = ce;

        const float E = E0 * (1.0f - 1e-4f * (T - 300.0f));

        // rate + analytic sensitivities: p = dgam/dov, q = d|gam|/dov, ag = |gam|
        auto eval = [&](const float* yv, float* r,
                        float& p, float& q, float& ag, float& gam) {
            float X = 0.0f;
#pragma unroll
            for (int i = 0; i < NHIST; ++i) X += yv[1 + i];
            const float ov  = yv[0] - X;
            const float f   = (__builtin_fabsf(ov) - s0) * inv_eta;
            const float sgn = (ov > 0.0f) ? 1.0f : ((ov < 0.0f) ? -1.0f : 0.0f);
            if (f > 0.0f) {
                // f^(n-1) via v_log_f32 / v_exp_f32; shared by rate (f^n) and Jacobian
                const float fn1 = __builtin_amdgcn_exp2f(nm1 * __builtin_amdgcn_logf(f));
                ag  = fn1 * f;
                gam = ag * sgn;
                p   = n * fn1 * inv_eta;
                q   = p * sgn;
            } else {
                p = 0.0f; q = 0.0f; ag = 0.0f; gam = 0.0f;
            }
            r[0] = E * (er - gam) * (1.0f - yv[7]);
#pragma unroll
            for (int i = 0; i < NHIST; ++i)
                r[1 + i] = C[i] * gam - g[i] * yv[1 + i] * ag;
            r[7] = A * ag;
        };

        float y0[NSIZE];
#pragma unroll
        for (int k = 0; k < NSIZE; ++k) y0[k] = y[k];

        // explicit predictor: y = y0 + dt * rate(y0)
        {
            float r[NSIZE]; float p, q, ag, gam;
            eval(y0, r, p, q, ag, gam);
#pragma unroll
            for (int k = 0; k < NSIZE; ++k) y[k] = y0[k] + dt * r[k];
        }

        // Newton: M = I - dt*J = B - U*V^T, B diagonal, rank(UV^T) = 2
        //   V1 = grad(ov) = (1,-1,...,-1,0),  V2 = e7 (damage column)
        // Exact solve via Sherman-Morrison-Woodbury (2x2 capacitance matrix).
#pragma unroll
        for (int it = 0; it < NEWTON_ITERS; ++it) {
            float r[NSIZE]; float p, q, ag, gam;
            eval(y, r, p, q, ag, gam);

            const float zs   = y[0] - y0[0] - dt * r[0];           // B0 = 1
            const float zd   = y[7] - y0[7] - dt * r[7];           // B7 = 1
            const float u1s  = -dt * E * (1.0f - y[7]) * p;        // dt*a, stress row
            const float u1d  =  dt * A * q;                        // dt*a, damage row
            const float c0dt = -dt * E * (er - gam);               // dt * dsdot/dd

            float q1  = zs;    // V1 . z
            float m11 = u1s;   // V1 . (B^-1 U1)
            float zh[NHIST], u1h[NHIST];
#pragma unroll
            for (int i = 0; i < NHIST; ++i) {
                const float Bi = frcp(1.0f + dt * g[i] * ag);      // B^-1 on history rows
                zh[i]  = (y[1 + i] - y0[1 + i] - dt * r[1 + i]) * Bi;
                u1h[i] = dt * (C[i] * p - g[i] * y[1 + i] * q) * Bi;
                q1  -= zh[i];
                m11 -= u1h[i];
            }
            const float q2   = zd;                                 // V2 . z
            const float idet = frcp((1.0f - m11) - c0dt * u1d);    // 2x2 det
            const float w1   = (q1 + c0dt * q2) * idet;
            const float w2   = (u1d * q1 + (1.0f - m11) * q2) * idet;

            y[0] -= zs + w1 * u1s + w2 * c0dt;
#pragma unroll
            for (int i = 0; i < NHIST; ++i) y[1 + i] -= zh[i] + w1 * u1h[i];
            y[7] -= zd + w1 * u1d;
        }

        // streaming non-temporal 2x b128 stores: wave covers 1 KB contiguous
        v4f lo = { y[0], y[1], y[2], y[3] };
        v4f hi = { y[4], y[5], y[6], y[7] };
        v4f* op = (v4f*)(out + ((size_t)t * (size_t)nbatch + (size_t)b) * NSIZE);
        __builtin_nontemporal_store(lo, op);
        __builtin_nontemporal_store(hi, op + 1);
    }
}

extern "C" void kernel_launch(void* const* d_in, const int* in_sizes, int n_in,
                              void* d_out, int out_size, void* d_ws, size_t ws_size,
                              hipStream_t stream) {
    (void)n_in; (void)out_size; (void)d_ws; (void)ws_size;
    const float* times   = (const float*)d_in[0];
    const float* strains = (const float*)d_in[1];
    const float* temps   = (const float*)d_in[2];
    const float* params  = (const float*)d_in[3];
    float* out = (float*)d_out;

    const int nbatch = in_sizes[0] / NTIME;           // times is (NTIME, nbatch)
    const int grid   = (nbatch + BLOCK - 1) / BLOCK;
    chaboche_wb_kernel<<<grid, BLOCK, 0, stream>>>(times, strains, temps, params, out, nbatch);
}